// RelationSpecificGNN_64871186038925
// MI455X (gfx1250) — compile-verified
//
#include <hip/hip_runtime.h>
#include <stdint.h>

#define N_NODES 100000   // NOTE: multiple of 16 -> no row guards needed in GEMM
#define N_EDGES 1600000
#define IN_DIM  128
#define HID_DIM 64

typedef float v2f __attribute__((ext_vector_type(2)));
typedef float v8f __attribute__((ext_vector_type(8)));

// ---------------------------------------------------------------- degree ----
__global__ void k_deg_init(float* __restrict__ deg) {
    int n = blockIdx.x * blockDim.x + threadIdx.x;
    if (n < N_NODES) deg[n] = 1.0f;   // self-loop contributes 1 to every node
}

__global__ void k_deg_count(const long long* __restrict__ ei,
                            float* __restrict__ deg) {
    int stride = gridDim.x * blockDim.x;
    for (int e = blockIdx.x * blockDim.x + threadIdx.x; e < N_EDGES; e += stride) {
        int d = (int)__builtin_nontemporal_load(&ei[N_EDGES + e]);  // dst row
        atomicAdd(&deg[d], 1.0f);
    }
}

__global__ void k_rsqrt_inplace(float* __restrict__ deg) {
    int n = blockIdx.x * blockDim.x + threadIdx.x;
    if (n < N_NODES) deg[n] = rsqrtf(deg[n]);  // deg >= 1 always
}

// ------------------------------------------------------- WMMA fp32 GEMM ----
// h[N,64] = x[N,128] @ W[128,64], exact fp32 via V_WMMA_F32_16X16X4_F32.
// Block = 128 threads = 4 waves; each wave computes rows [m0, m0+16) x all 64
// cols with 4 accumulators (8 VGPRs each). W is staged into LDS *pair-
// interleaved* so each lane's (K, K+1) B-fragment is one aligned ds_load_b64:
//   sW2[p*128 + 2n + 0] = W[2p  ][n]
//   sW2[p*128 + 2n + 1] = W[2p+1][n]      (p = K/2, 64 pair-rows)
// N_NODES % 16 == 0, so every live wave owns a fully in-range 16-row tile.
__global__ __launch_bounds__(128)
void k_gemm_wmma(const float* __restrict__ x,
                 const float* __restrict__ W,
                 float* __restrict__ h) {
    __shared__ float sW2[(IN_DIM / 2) * (HID_DIM * 2)];   // 8192 floats = 32 KB

    // stage + interleave: i enumerates (pair-row p, col n)
    for (int i = threadIdx.x; i < (IN_DIM / 2) * HID_DIM; i += blockDim.x) {
        int p = i >> 6;          // 0..63
        int n = i & 63;          // 0..63
        float2 w2;
        w2.x = W[(2 * p)     * HID_DIM + n];
        w2.y = W[(2 * p + 1) * HID_DIM + n];
        ((float2*)sW2)[p * HID_DIM + n] = w2;
    }
    __syncthreads();

    const int lane = threadIdx.x & 31;
    const int wave = threadIdx.x >> 5;
    const int m0   = (blockIdx.x * 4 + wave) * 16;
    if (m0 >= N_NODES) return;                 // wave-uniform exit (EXEC all-1)

    const int hi = lane >> 4;                  // 0: lanes 0-15, 1: lanes 16-31
    const int lm = lane & 15;

    const float* __restrict__ xr  = x + (long)(m0 + lm) * IN_DIM;
    const v2f*   __restrict__ sWp = (const v2f*)sW2;   // [64 pair-rows][64 cols]

    v8f acc0 = {}, acc1 = {}, acc2 = {}, acc3 = {};

    #pragma unroll 4
    for (int k0 = 0; k0 < IN_DIM; k0 += 4) {
        // A 16x4 f32 frag: lanes 0-15 carry K=k0,k0+1; lanes 16-31 K=k0+2,k0+3
        v2f a = *(const v2f*)(xr + k0 + 2 * hi);

        // B 4x16 frags: pair-row index = k0/2 + hi, one b64 load per n-tile
        const v2f* wrow = sWp + (k0 / 2 + hi) * HID_DIM + lm;
        v2f b0 = wrow[0];
        v2f b1 = wrow[16];
        v2f b2 = wrow[32];
        v2f b3 = wrow[48];

        acc0 = __builtin_amdgcn_wmma_f32_16x16x4_f32(false, a, false, b0, (short)0, acc0, false, false);
        acc1 = __builtin_amdgcn_wmma_f32_16x16x4_f32(false, a, false, b1, (short)0, acc1, false, false);
        acc2 = __builtin_amdgcn_wmma_f32_16x16x4_f32(false, a, false, b2, (short)0, acc2, false, false);
        acc3 = __builtin_amdgcn_wmma_f32_16x16x4_f32(false, a, false, b3, (short)0, acc3, false, false);
    }

    // C/D layout: VGPR v holds row m0+v (lanes 0-15) / m0+v+8 (lanes 16-31);
    // all rows in range (N_NODES % 16 == 0) -> unguarded stores.
    #pragma unroll
    for (int v = 0; v < 8; ++v) {
        float* hr = h + (long)(m0 + v + 8 * hi) * HID_DIM + lm;
        hr[0]  = acc0[v];
        hr[16] = acc1[v];
        hr[32] = acc2[v];
        hr[48] = acc3[v];
    }
}

// ----------------------------------------------- out = b + self-loop term ---
__global__ void k_out_init(const float* __restrict__ h,
                           const float* __restrict__ dinv,
                           const float* __restrict__ bias,
                           float* __restrict__ out) {
    long i = (long)blockIdx.x * blockDim.x + threadIdx.x;
    if (i < (long)N_NODES * HID_DIM) {
        int n = (int)(i >> 6);
        int c = (int)(i & 63);
        float di = dinv[n];
        out[i] = bias[c] + h[i] * di * di;     // self-loop: norm = dinv[n]^2
    }
}

// ------------------------------------------------------- edge scatter-add ---
// One wave per edge: 32 lanes x 2 columns; gather h[src] (L2-resident),
// atomic-add into out[dst] (RMW resolves in the 192 MB L2 on MI455X).
__global__ __launch_bounds__(256)
void k_agg(const long long* __restrict__ ei,
           const float* __restrict__ h,
           const float* __restrict__ dinv,
           float* __restrict__ out) {
    int  lane = threadIdx.x & 31;
    long gw   = ((long)blockIdx.x * blockDim.x + threadIdx.x) >> 5;
    long nw   = ((long)gridDim.x * blockDim.x) >> 5;
    for (long e = gw; e < N_EDGES; e += nw) {
        int s = (int)__builtin_nontemporal_load(&ei[e]);            // src
        int d = (int)__builtin_nontemporal_load(&ei[N_EDGES + e]);  // dst
        float nrm = dinv[s] * dinv[d];
        const float* __restrict__ hs = h + (long)s * HID_DIM;
        float* __restrict__ od = out + (long)d * HID_DIM;
        atomicAdd(&od[lane],      hs[lane]      * nrm);
        atomicAdd(&od[lane + 32], hs[lane + 32] * nrm);
    }
}

__global__ void k_relu(float* __restrict__ out) {
    long i = (long)blockIdx.x * blockDim.x + threadIdx.x;
    if (i < (long)N_NODES * HID_DIM) out[i] = fmaxf(out[i], 0.0f);
}

// ------------------------------------------------------------------ launch --
extern "C" void kernel_launch(void* const* d_in, const int* in_sizes, int n_in,
                              void* d_out, int out_size, void* d_ws, size_t ws_size,
                              hipStream_t stream) {
    const float*     x  = (const float*)d_in[0];
    const long long* ei = (const long long*)d_in[1];   // int64 [2, E]
    const float*     W  = (const float*)d_in[2];
    const float*     b  = (const float*)d_in[3];
    float* out = (float*)d_out;

    // Workspace layout: h [N*64 f32] then deg/dinv [N f32]  (~26 MB)
    float* h   = (float*)d_ws;
    float* deg = h + (size_t)N_NODES * HID_DIM;

    const long OUT_ELEMS = (long)N_NODES * HID_DIM;

    k_deg_init     <<<(N_NODES + 255) / 256, 256, 0, stream>>>(deg);
    k_deg_count    <<<2048, 256, 0, stream>>>(ei, deg);
    k_rsqrt_inplace<<<(N_NODES + 255) / 256, 256, 0, stream>>>(deg);

    k_gemm_wmma    <<<(N_NODES + 63) / 64, 128, 0, stream>>>(x, W, h);

    k_out_init     <<<(OUT_ELEMS + 255) / 256, 256, 0, stream>>>(h, deg, b, out);
    k_agg          <<<4096, 256, 0, stream>>>(ei, h, deg, out);
    k_relu         <<<(OUT_ELEMS + 255) / 256, 256, 0, stream>>>(out);
}